// EssentialMSAC_32358283608477
// MI455X (gfx1250) — compile-verified
//
#include <hip/hip_runtime.h>

typedef __attribute__((ext_vector_type(2))) float v2f;
typedef __attribute__((ext_vector_type(8))) float v8f;

#define MCHUNK 128
#define WAVES_PER_BLOCK 8
#define PTS_PER_WAVE 32   // two 16-point WMMA tiles per wave

// D = A(16x4) x B(4x16) + C ; 16 points x 2 models per WMMA, 2 tiles per wave.
// Channels (rows of A / D), per model:
//   ch0: u0 = E00*x1 + E01*y1             + E02 (const via C)
//   ch1: u1 = E10*x1 + E11*y1             + E12
//   ch2: v0 =               E00*x2+E10*y2 + E20
//   ch3: v1 =               E01*x2+E11*y2 + E21
//   ch4: v2 =               E02*x2+E12*y2 + E22
// JJ = u0^2+u1^2+v0^2+v1^2 ; num = x1*v0 + y1*v1 + v2
// out = 1 - clip((num/thr)^2 * rcp(JJ), 0, 1)
__global__ __launch_bounds__(256) void essential_msac_wmma(
    const float* __restrict__ models, const float* __restrict__ points,
    const float* __restrict__ K1, const float* __restrict__ K2,
    float* __restrict__ out, int B, int N)
{
    __shared__ alignas(16) float2 sCoef[MCHUNK * 16];   // 16 KB
    __shared__ alignas(16) float  sConst[MCHUNK * 8];   //  4 KB

    const int lane = threadIdx.x & 31;
    const int wave = threadIdx.x >> 5;
    const int col  = lane & 15;   // point within tile / channel row
    const int h    = lane >> 4;   // half-wave: K-pair (A) / model (C,D)
    const int pi   = col & 7;     // channel index within a model

    const int  pbase = (blockIdx.x * WAVES_PER_BLOCK + wave) * PTS_PER_WAVE;
    const int  p0 = pbase + col;        // tile 0
    const int  p1 = pbase + 16 + col;   // tile 1
    const bool v0ok = (p0 < N);
    const bool v1ok = (p1 < N);
    const int  pc0 = v0ok ? p0 : (N - 1);
    const int  pc1 = v1ok ? p1 : (N - 1);

    const float4 ptA = reinterpret_cast<const float4*>(points)[pc0];
    const float4 ptB = reinterpret_cast<const float4*>(points)[pc1];
    const float x1a = ptA.x, y1a = ptA.y;
    const float x1b = ptB.x, y1b = ptB.y;

    // B operands (4x16): lanes 0-15 rows K=0,1 (x1,y1); lanes 16-31 rows K=2,3 (x2,y2)
    v2f bmat0, bmat1;
    bmat0.x = h ? ptA.z : ptA.x;  bmat0.y = h ? ptA.w : ptA.y;
    bmat1.x = h ? ptB.z : ptB.x;  bmat1.y = h ? ptB.w : ptB.y;

    const float f1  = (K1[0] + K1[4]) * 0.5f;
    const float f2  = (K2[0] + K2[4]) * 0.5f;
    const float thr = (1.0f / f1 + 1.0f / f2) * 0.5f;
    const float inv_thr = 1.0f / thr;

    const int coefLaneOff  = ((col < 8) ? 0 : 16) + pi * 2 + h; // col>=8 -> model mp+1
    const int constLaneOff = h * 8;                              // h -> model mp+h

    // fast epilogue: single v_rcp instead of IEEE divide; clamp folds into mul
    auto epi = [&](const v8f& d, float px, float py) -> float {
        const float u0 = d[0], u1 = d[1], w0 = d[2], w1 = d[3], w2 = d[4];
        const float JJ  = u0 * u0 + u1 * u1 + w0 * w0 + w1 * w1;
        const float num = (w0 * px + w1 * py + w2) * inv_thr;
        const float ratio = num * num * __builtin_amdgcn_rcpf(JJ);
        return 1.0f - fminf(fmaxf(ratio, 0.0f), 1.0f);
    };

    for (int c0 = 0; c0 < B; c0 += MCHUNK) {
        const int cmax = (B - c0 < MCHUNK) ? (B - c0) : MCHUNK;
        __syncthreads();
        // Build E = skew(t) @ R and A/C tables for this chunk of models.
        for (int m = threadIdx.x; m < cmax; m += blockDim.x) {
            const float* Mp = models + (size_t)(c0 + m) * 12;
            const float R00=Mp[0], R01=Mp[1], R02=Mp[2],  t0=Mp[3];
            const float R10=Mp[4], R11=Mp[5], R12=Mp[6],  t1=Mp[7];
            const float R20=Mp[8], R21=Mp[9], R22=Mp[10], t2=Mp[11];
            const float E00=t1*R20-t2*R10, E01=t1*R21-t2*R11, E02=t1*R22-t2*R12;
            const float E10=t2*R00-t0*R20, E11=t2*R01-t0*R21, E12=t2*R02-t0*R22;
            const float E20=t0*R10-t1*R00, E21=t0*R11-t1*R01, E22=t0*R12-t1*R02;
            float2* C = sCoef + m * 16;
            C[0]  = make_float2(E00, E01); C[1]  = make_float2(0.f, 0.f);   // u0
            C[2]  = make_float2(E10, E11); C[3]  = make_float2(0.f, 0.f);   // u1
            C[4]  = make_float2(0.f, 0.f); C[5]  = make_float2(E00, E10);   // v0
            C[6]  = make_float2(0.f, 0.f); C[7]  = make_float2(E01, E11);   // v1
            C[8]  = make_float2(0.f, 0.f); C[9]  = make_float2(E02, E12);   // v2
            C[10] = C[11] = C[12] = C[13] = C[14] = C[15] = make_float2(0.f, 0.f);
            float* Kc = sConst + m * 8;
            Kc[0]=E02; Kc[1]=E12; Kc[2]=E20; Kc[3]=E21; Kc[4]=E22;
            Kc[5]=0.f; Kc[6]=0.f; Kc[7]=0.f;
        }
        __syncthreads();

        const int mpairs = cmax & ~1;   // full model pairs (always == cmax for B=512)
        for (int mp = 0; mp < mpairs; mp += 2) {
            const float2 cf = sCoef[mp * 16 + coefLaneOff];
            v2f amat; amat.x = cf.x; amat.y = cf.y;

            const float4* kc = reinterpret_cast<const float4*>(sConst + mp * 8 + constLaneOff);
            const float4 k0 = kc[0], k1 = kc[1];
            v8f cmat = {k0.x, k0.y, k0.z, k0.w, k1.x, k1.y, k1.z, k1.w};

            // Two tiles share one A/C load; WMMAs pipeline back-to-back.
            v8f d0 = __builtin_amdgcn_wmma_f32_16x16x4_f32(
                false, amat, false, bmat0, (short)0, cmat, false, false);
            v8f d1 = __builtin_amdgcn_wmma_f32_16x16x4_f32(
                false, amat, false, bmat1, (short)0, cmat, false, false);

            const float r0 = epi(d0, x1a, y1a);
            const float r1 = epi(d1, x1b, y1b);

            const size_t row = (size_t)(c0 + mp + h) * N;
            if (v0ok) out[row + p0] = r0;
            if (v1ok) out[row + p1] = r1;
        }
        // Generality tail: odd chunk -> last model handled by h==0 lanes only.
        if (mpairs < cmax) {
            const int mp = mpairs;
            const float2 cf = sCoef[mp * 16 + coefLaneOff];
            v2f amat; amat.x = (col < 8) ? cf.x : 0.f; amat.y = (col < 8) ? cf.y : 0.f;
            const float4* kc = reinterpret_cast<const float4*>(sConst + mp * 8);
            const float4 k0 = kc[0], k1 = kc[1];
            v8f cmat = {k0.x, k0.y, k0.z, k0.w, k1.x, k1.y, k1.z, k1.w};
            v8f d0 = __builtin_amdgcn_wmma_f32_16x16x4_f32(
                false, amat, false, bmat0, (short)0, cmat, false, false);
            v8f d1 = __builtin_amdgcn_wmma_f32_16x16x4_f32(
                false, amat, false, bmat1, (short)0, cmat, false, false);
            const float r0 = epi(d0, x1a, y1a);
            const float r1 = epi(d1, x1b, y1b);
            const size_t row = (size_t)(c0 + mp) * N;
            if (v0ok && h == 0) out[row + p0] = r0;
            if (v1ok && h == 0) out[row + p1] = r1;
        }
    }
}

extern "C" void kernel_launch(void* const* d_in, const int* in_sizes, int n_in,
                              void* d_out, int out_size, void* d_ws, size_t ws_size,
                              hipStream_t stream) {
    const float* models = (const float*)d_in[0];   // (B,3,4)
    const float* points = (const float*)d_in[1];   // (N,4)
    const float* K1     = (const float*)d_in[2];   // (3,3)
    const float* K2     = (const float*)d_in[3];   // (3,3)
    float* out = (float*)d_out;                    // (B,N)

    const int B = in_sizes[0] / 12;
    const int N = in_sizes[1] / 4;

    const int ptsPerBlock = WAVES_PER_BLOCK * PTS_PER_WAVE;  // 256
    const int grid = (N + ptsPerBlock - 1) / ptsPerBlock;
    essential_msac_wmma<<<grid, 256, 0, stream>>>(models, points, K1, K2, out, B, N);
}